// SparseInverseConv3d_39393440038986
// MI455X (gfx1250) — compile-verified
//
#include <hip/hip_runtime.h>

typedef float v2f __attribute__((ext_vector_type(2)));
typedef float v8f __attribute__((ext_vector_type(8)));

// Problem sizes are fixed by the reference.
constexpr int K_OFF = 27;
constexpr int P_PAIRS = 100000;
constexpr int C_IN = 32;
constexpr int C_OUT = 32;

constexpr int TILES_PER_CHUNK = 16;                 // 16 row-tiles (256 rows) per wave task
constexpr int ROWS_PER_CHUNK = TILES_PER_CHUNK * 16;
constexpr int CHUNKS_PER_K = (P_PAIRS + ROWS_PER_CHUNK - 1) / ROWS_PER_CHUNK; // 391
constexpr int WAVES_PER_BLOCK = 8;                  // 256 threads, wave32
constexpr int TOTAL_WAVE_TASKS = K_OFF * CHUNKS_PER_K; // 10557

// out[i] = bias[i % 32], vectorized as float4 (32 % 4 == 0).
__global__ void bias_init_kernel(const float* __restrict__ bias,
                                 float* __restrict__ out, int nelem4) {
  int i = blockIdx.x * blockDim.x + threadIdx.x;
  if (i >= nelem4) return;
  const float4* b4 = reinterpret_cast<const float4*>(bias);
  reinterpret_cast<float4*>(out)[i] = b4[i & 7];
}

__global__ __launch_bounds__(256) void sparse_conv_wmma_kernel(
    const float* __restrict__ feats,       // [N_IN, 32]
    const float* __restrict__ weight,      // [27, 32, 32]
    const long long* __restrict__ pairs_in,  // [27, P] int64
    const long long* __restrict__ pairs_out, // [27, P] int64
    float* __restrict__ out)               // [N_OUT, 32], pre-set to bias
{
  const int wave = blockIdx.x * WAVES_PER_BLOCK + (threadIdx.x >> 5);
  if (wave >= TOTAL_WAVE_TASKS) return;    // wave-uniform: EXEC stays all-ones

  const int k     = wave / CHUNKS_PER_K;
  const int chunk = wave % CHUNKS_PER_K;
  const int lane  = threadIdx.x & 31;
  const int g     = lane >> 4;             // half-wave group (K-pair / row-half select)
  const int n     = lane & 15;             // column within 16-wide tile; also A-row id

  // ---- Load B fragments for weight[k] (32x32) once per wave task ----
  // B 4x16 fp32 layout: lane (g,n) holds {W[2g][n], W[2g+1][n]} of each 4-row K slice.
  const float* wk = weight + k * (C_IN * C_OUT) + (2 * g) * C_OUT + n;
  v2f b0[8], b1[8];
#pragma unroll
  for (int s = 0; s < 8; ++s) {
    const float* wp = wk + s * 4 * C_OUT;
    b0[s].x = wp[0];            // K=4s+2g,   N=n
    b0[s].y = wp[C_OUT];        // K=4s+2g+1, N=n
    b1[s].x = wp[16];           // N=16+n
    b1[s].y = wp[C_OUT + 16];
  }

  const long long* pin  = pairs_in  + (long long)k * P_PAIRS;
  const long long* pout = pairs_out + (long long)k * P_PAIRS;

  const int row0 = chunk * ROWS_PER_CHUNK;
  for (int t = 0; t < TILES_PER_CHUNK; ++t) {
    const int row_base = row0 + t * 16;
    if (row_base >= P_PAIRS) break;        // P % 16 == 0, so tiles are always full

    // Gather: lane's A-row is rulebook row (row_base + n); lanes 0-15 read
    // columns {4s+0,4s+1}, lanes 16-31 read {4s+2,4s+3} -> row read exactly once.
    const long long idx_in = pin[row_base + n];
    const float* arow = feats + idx_in * C_IN + 2 * g;

    v8f acc0 = {};
    v8f acc1 = {};
#pragma unroll
    for (int s = 0; s < 8; ++s) {
      v2f a = *reinterpret_cast<const v2f*>(arow + 4 * s);  // 8B-aligned
      // 8 args: (neg_a, A, neg_b, B, c_mod, C, reuse_a, reuse_b)
      acc0 = __builtin_amdgcn_wmma_f32_16x16x4_f32(false, a, false, b0[s],
                                                   (short)0, acc0, false, false);
      acc1 = __builtin_amdgcn_wmma_f32_16x16x4_f32(false, a, false, b1[s],
                                                   (short)0, acc1, false, false);
    }

    // Scatter-add. D layout: VGPR r -> row m = r + 8g, col n (acc0) / 16+n (acc1).
#pragma unroll
    for (int r = 0; r < 8; ++r) {
      const long long o = pout[row_base + r + 8 * g];
      float* orow = out + o * C_OUT + n;
      unsafeAtomicAdd(orow,      acc0[r]);   // global_atomic_add_f32
      unsafeAtomicAdd(orow + 16, acc1[r]);
    }
  }
}

extern "C" void kernel_launch(void* const* d_in, const int* in_sizes, int n_in,
                              void* d_out, int out_size, void* d_ws, size_t ws_size,
                              hipStream_t stream) {
  const float*     feats     = (const float*)d_in[0];
  const float*     weight    = (const float*)d_in[1];
  const float*     bias      = (const float*)d_in[2];
  const long long* pairs_in  = (const long long*)d_in[3];
  const long long* pairs_out = (const long long*)d_in[4];
  float* out = (float*)d_out;

  // 1) out = bias (broadcast), float4-vectorized.
  const int nelem4 = out_size / 4;
  bias_init_kernel<<<(nelem4 + 255) / 256, 256, 0, stream>>>(bias, out, nelem4);

  // 2) gather -> WMMA fp32 GEMM -> scatter-add.
  const int blocks = (TOTAL_WAVE_TASKS + WAVES_PER_BLOCK - 1) / WAVES_PER_BLOCK;
  sparse_conv_wmma_kernel<<<blocks, 256, 0, stream>>>(feats, weight, pairs_in,
                                                      pairs_out, out);
}